// Shrinkagev3p_78245714198911
// MI455X (gfx1250) — compile-verified
//
#include <hip/hip_runtime.h>

typedef __attribute__((ext_vector_type(2))) float v2f;
typedef __attribute__((ext_vector_type(4))) float f4;
typedef __attribute__((ext_vector_type(8))) float v8f;

constexpr int Bn = 32;      // batch
constexpr int Cn = 512;     // channels
constexpr int Ln = 8192;    // length
constexpr float BN_EPS = 1e-5f;

// ---------------------------------------------------------------------------
// Kernel 1: avg[b,c] = mean(x[b,c,:])  — one block per (b,c) row.
// Streaming read of x (no reuse) -> non-temporal loads.
// ---------------------------------------------------------------------------
__global__ void __launch_bounds__(256)
avg_kernel(const float* __restrict__ x, float* __restrict__ avg) {
    __shared__ float red[256];
    const int row = blockIdx.x;                         // b*Cn + c
    const f4* xr = (const f4*)(x + (size_t)row * Ln);
    float s = 0.0f;
    #pragma unroll
    for (int i = threadIdx.x; i < Ln / 4; i += 256) {   // 8 float4 per thread
        f4 v = __builtin_nontemporal_load(&xr[i]);
        s += v.x + v.y + v.z + v.w;
    }
    red[threadIdx.x] = s;
    __syncthreads();
    for (int off = 128; off > 0; off >>= 1) {
        if (threadIdx.x < off) red[threadIdx.x] += red[threadIdx.x + off];
        __syncthreads();
    }
    if (threadIdx.x == 0) avg[row] = red[0] * (1.0f / (float)Ln);
}

// ---------------------------------------------------------------------------
// Kernel 2: the gate MLP, single workgroup, fp32 WMMA (V_WMMA_F32_16X16X4_F32).
//   h   = avg @ W1^T + b1 ; BN(eval) ; ReLU        (GEMM1, result in LDS)
//   gate= sigmoid(h @ W2^T + b2)                   (GEMM2)
//   thr = avg * gate
// A-fragment (16x4 f32): lane<16 holds M=lane, K={k,k+1}; lane>=16 K={k+2,k+3}.
// B-fragment (4x16) mirrors with N=lane&15. C/D: VGPR r -> M = r + 8*(lane>>4),
// N = lane&15.
// ---------------------------------------------------------------------------
__global__ void __launch_bounds__(512)
gate_kernel(const float* __restrict__ avg,
            const float* __restrict__ W1, const float* __restrict__ b1,
            const float* __restrict__ gamma, const float* __restrict__ beta,
            const float* __restrict__ mean,  const float* __restrict__ var,
            const float* __restrict__ W2, const float* __restrict__ b2,
            float* __restrict__ thr) {
    __shared__ float hls[Bn * Cn];                      // 64 KB
    const int lane = threadIdx.x & 31;
    const int wv   = threadIdx.x >> 5;                  // 16 waves
    const int lm   = lane & 15;
    const int hi   = lane >> 4;                         // 0/1 -> K pair select
    const int koff = 2 * hi;

    // ---- GEMM1: 64 tiles (2 M-tiles x 32 N-tiles), K = 512 in steps of 4 ----
    for (int t = wv; t < 64; t += 16) {
        const int m0 = (t & 1) * 16;
        const int n0 = (t >> 1) * 16;
        const int m  = m0 + lm;                          // A row for this lane
        const int n  = n0 + lm;                          // B col for this lane
        const float* arow = avg + m * Cn;
        const float* brow = W1  + n * Cn;                // B[k][n] = W1[n][k]
        v8f acc = {};
        for (int k = 0; k < Cn; k += 4) {
            v2f a = { arow[k + koff], arow[k + koff + 1] };
            v2f b = { brow[k + koff], brow[k + koff + 1] };
            acc = __builtin_amdgcn_wmma_f32_16x16x4_f32(
                false, a, false, b, (short)0, acc, false, false);
        }
        // bias + BatchNorm(eval) + ReLU, write to LDS
        const float inv = rsqrtf(var[n] + BN_EPS);
        const float ga = gamma[n], be = beta[n], mu = mean[n], bb = b1[n];
        #pragma unroll
        for (int r = 0; r < 8; ++r) {
            const int mr = m0 + r + 8 * hi;              // C/D row
            float h = acc[r] + bb;
            h = ga * (h - mu) * inv + be;
            hls[mr * Cn + n] = fmaxf(h, 0.0f);
        }
    }
    __syncthreads();

    // ---- GEMM2: A from LDS, then sigmoid, thr = avg * gate ----
    for (int t = wv; t < 64; t += 16) {
        const int m0 = (t & 1) * 16;
        const int n0 = (t >> 1) * 16;
        const int m  = m0 + lm;
        const int n  = n0 + lm;
        const float* arow = hls + m * Cn;
        const float* brow = W2  + n * Cn;
        v8f acc = {};
        for (int k = 0; k < Cn; k += 4) {
            v2f a = { arow[k + koff], arow[k + koff + 1] };
            v2f b = { brow[k + koff], brow[k + koff + 1] };
            acc = __builtin_amdgcn_wmma_f32_16x16x4_f32(
                false, a, false, b, (short)0, acc, false, false);
        }
        const float bb = b2[n];
        #pragma unroll
        for (int r = 0; r < 8; ++r) {
            const int mr = m0 + r + 8 * hi;
            const float z = acc[r] + bb;
            const float gate = 1.0f / (1.0f + expf(-z));
            thr[mr * Cn + n] = avg[mr * Cn + n] * gate;
        }
    }
}

// ---------------------------------------------------------------------------
// Kernel 3: out = min(|x| - thr[b,c], 0) — streaming, float4, NT load/store.
// grid = (Ln/4/256, Bn*Cn)
// ---------------------------------------------------------------------------
__global__ void __launch_bounds__(256)
shrink_kernel(const float* __restrict__ x, const float* __restrict__ thr,
              float* __restrict__ out) {
    const int row = blockIdx.y;                          // b*Cn + c
    const int i   = blockIdx.x * 256 + threadIdx.x;      // float4 index in row
    const float t = thr[row];
    const f4* xr = (const f4*)(x   + (size_t)row * Ln);
    f4*       orw = (f4*)(out + (size_t)row * Ln);
    f4 v = __builtin_nontemporal_load(&xr[i]);
    f4 r;
    r.x = fminf(fabsf(v.x) - t, 0.0f);
    r.y = fminf(fabsf(v.y) - t, 0.0f);
    r.z = fminf(fabsf(v.z) - t, 0.0f);
    r.w = fminf(fabsf(v.w) - t, 0.0f);
    __builtin_nontemporal_store(r, &orw[i]);
}

// ---------------------------------------------------------------------------
extern "C" void kernel_launch(void* const* d_in, const int* in_sizes, int n_in,
                              void* d_out, int out_size, void* d_ws, size_t ws_size,
                              hipStream_t stream) {
    const float* x     = (const float*)d_in[0];
    const float* W1    = (const float*)d_in[1];
    const float* b1    = (const float*)d_in[2];
    const float* gamma = (const float*)d_in[3];
    const float* beta  = (const float*)d_in[4];
    const float* mean  = (const float*)d_in[5];
    const float* var   = (const float*)d_in[6];
    const float* W2    = (const float*)d_in[7];
    const float* b2    = (const float*)d_in[8];
    float* out = (float*)d_out;

    float* avg = (float*)d_ws;            // 32*512 floats
    float* thr = avg + Bn * Cn;           // 32*512 floats

    avg_kernel<<<Bn * Cn, 256, 0, stream>>>(x, avg);
    gate_kernel<<<1, 512, 0, stream>>>(avg, W1, b1, gamma, beta, mean, var,
                                       W2, b2, thr);
    dim3 g3(Ln / 4 / 256, Bn * Cn);
    shrink_kernel<<<g3, 256, 0, stream>>>(x, thr, out);
}